// Linear_15874199126073
// MI455X (gfx1250) — compile-verified
//
#include <hip/hip_runtime.h>

typedef __attribute__((ext_vector_type(2))) float v2f;
typedef __attribute__((ext_vector_type(4))) float v4f;
typedef __attribute__((ext_vector_type(8))) float v8f;

#define N_NODES        50000
#define FEAT_DIM       2048
#define MUL            128
#define LDS_STRIDE     132     // 128 + 4 floats pad: kills LDS bank conflicts
#define TILE_ROWS      128
#define BLOCK_THREADS  128     // 4 waves of 32; each wave computes 32 rows x 128 cols

// D = irrep dimension, XOFF = feature/out column offset, WOFF = weight offset.
template <int D, int XOFF, int WOFF, bool HAS_BIAS>
__global__ void __launch_bounds__(BLOCK_THREADS)
equi_linear_block(const float* __restrict__ features,
                  const float* __restrict__ weight,
                  const float* __restrict__ bias,
                  float* __restrict__ out,
                  int ntiles)
{
    extern __shared__ float smem[];
    float* Wlds = smem;                          // [128][LDS_STRIDE]  (u-major)
    float* Flds = smem + MUL * LDS_STRIDE;       // [TILE_ROWS][LDS_STRIDE]

    const int tid  = threadIdx.x;
    const int lane = tid & 31;
    const int wv   = tid >> 5;
    const int l16  = lane & 15;
    const int h    = lane >> 4;        // which half of the wave (K/M split)
    const int R    = N_NODES * D;      // logical GEMM rows for this block

    // ---- stage W block (128x128 f32 = 64KB) into LDS once, coalesced ----
    for (int e = tid; e < MUL * MUL; e += BLOCK_THREADS) {
        int u = e >> 7;
        int c = e & 127;
        Wlds[u * LDS_STRIDE + c] = weight[WOFF + e];
    }

    const float pw = 0.08838834764831845f;   // 1/sqrt(128) path normalization

    for (int tile = blockIdx.x; tile < ntiles; tile += gridDim.x) {
        const int r0 = tile * TILE_ROWS;

        __syncthreads();   // previous tile's compute done (also fences W load)

        // ---- stage 128-row x 128-u feature tile into LDS ----
        // Row r = n*D + i reads features[n*2048 + XOFF + u*D + i]; per node the
        // whole (u,i) block is contiguous, so these loads hit full cachelines.
        for (int e4 = tid; e4 < TILE_ROWS * (MUL / 4); e4 += BLOCK_THREADS) {
            int row = e4 >> 5;              // MUL/4 == 32 vec4 per row
            int u4  = (e4 & 31) << 2;
            int r   = r0 + row;
            if (r >= R) r = R - 1;          // clamp: tail rows load valid data
            int n = r / D;
            int i = r - n * D;
            const float* g = features + n * FEAT_DIM + XOFF + i + u4 * D;
            v4f v;
            if (D == 1) {
                v = *(const v4f*)g;         // fully coalesced b128
            } else {
                v.x = g[0]; v.y = g[D]; v.z = g[2 * D]; v.w = g[3 * D];
            }
            *(v4f*)(Flds + row * LDS_STRIDE + u4) = v;
        }
        __syncthreads();

        // ---- WMMA: wave owns rows [wv*32, wv*32+32), all 128 output cols ----
        v8f acc[2][8] = {};
        const int rw = wv * 32;

        for (int u0 = 0; u0 < MUL; u0 += 4) {
            // A fragments (16x4 f32): lane l16 holds row M=l16,
            // vgpr0/1 = K = 2h / 2h+1  -> contiguous ds_load_b64 from Flds.
            v2f a0 = *(const v2f*)(Flds + (rw +      l16) * LDS_STRIDE + u0 + 2 * h);
            v2f a1 = *(const v2f*)(Flds + (rw + 16 + l16) * LDS_STRIDE + u0 + 2 * h);
#pragma unroll
            for (int wt = 0; wt < 8; ++wt) {
                // B fragment (4x16 f32): lane l16 holds col N=l16,
                // vgpr0/1 = K = 2h / 2h+1 -> rows of W in LDS.
                v2f b;
                b.x = Wlds[(u0 + 2 * h    ) * LDS_STRIDE + wt * 16 + l16];
                b.y = Wlds[(u0 + 2 * h + 1) * LDS_STRIDE + wt * 16 + l16];
                acc[0][wt] = __builtin_amdgcn_wmma_f32_16x16x4_f32(
                    false, a0, false, b, (short)0, acc[0][wt], false, false);
                acc[1][wt] = __builtin_amdgcn_wmma_f32_16x16x4_f32(
                    false, a1, false, b, (short)0, acc[1][wt], false, false);
            }
        }

        // ---- epilogue: scale, bias (scalar block), predicated scatter store ----
#pragma unroll
        for (int wt = 0; wt < 8; ++wt) {
            const int col = wt * 16 + l16;          // output channel w
            float bv = 0.0f;
            if (HAS_BIAS) bv = bias[col];
#pragma unroll
            for (int mt = 0; mt < 2; ++mt) {
#pragma unroll
                for (int j = 0; j < 8; ++j) {
                    // C/D layout: vgpr j -> M = j (lanes 0-15) / j+8 (lanes 16-31)
                    int r = r0 + rw + mt * 16 + j + 8 * h;
                    if (r < R) {
                        int n = r / D;
                        int i = r - n * D;
                        out[n * FEAT_DIM + XOFF + col * D + i] =
                            acc[mt][wt][j] * pw + bv;
                    }
                }
            }
        }
    }
}

static inline int grid_for(int d) {
    int ntiles = (N_NODES * d + TILE_ROWS - 1) / TILE_ROWS;
    return ntiles < 1024 ? ntiles : 1024;
}

extern "C" void kernel_launch(void* const* d_in, const int* in_sizes, int n_in,
                              void* d_out, int out_size, void* d_ws, size_t ws_size,
                              hipStream_t stream) {
    (void)in_sizes; (void)n_in; (void)out_size; (void)d_ws; (void)ws_size;

    const float* features = (const float*)d_in[0];
    const float* weight   = (const float*)d_in[1];
    const float* bias     = (const float*)d_in[2];
    float*       out      = (float*)d_out;

    const size_t shmem = (size_t)(MUL + TILE_ROWS) * LDS_STRIDE * sizeof(float);

    {   // block 0: 128x0e (d=1), gets bias
        int nt = (N_NODES * 1 + TILE_ROWS - 1) / TILE_ROWS;
        equi_linear_block<1, 0, 0, true>
            <<<grid_for(1), BLOCK_THREADS, shmem, stream>>>(features, weight, bias, out, nt);
    }
    {   // block 1: 128x1o (d=3)
        int nt = (N_NODES * 3 + TILE_ROWS - 1) / TILE_ROWS;
        equi_linear_block<3, 128, 16384, false>
            <<<grid_for(3), BLOCK_THREADS, shmem, stream>>>(features, weight, bias, out, nt);
    }
    {   // block 2: 128x2e (d=5)
        int nt = (N_NODES * 5 + TILE_ROWS - 1) / TILE_ROWS;
        equi_linear_block<5, 512, 32768, false>
            <<<grid_for(5), BLOCK_THREADS, shmem, stream>>>(features, weight, bias, out, nt);
    }
    {   // block 3: 128x3o (d=7)
        int nt = (N_NODES * 7 + TILE_ROWS - 1) / TILE_ROWS;
        equi_linear_block<7, 1152, 49152, false>
            <<<grid_for(7), BLOCK_THREADS, shmem, stream>>>(features, weight, bias, out, nt);
    }
}